// CTCLoss_61280593379969
// MI455X (gfx1250) — compile-verified
//
#include <hip/hip_runtime.h>
#include <stdint.h>

// ---- problem constants (match reference setup_inputs) ----
#define TT    1000
#define BB    64
#define VV    512
#define SS    100
#define SE    201          // 2*S+1 extended positions
#define NPOS  7            // positions per lane (7*32 = 224 >= 201)
#define ROWF  224          // buffered floats per timestep (padded)
#define DEPTH 8            // async pipeline depth (power of 2)
#define BLANKTOK 1
#define NEGV  (-1e30f)

// ---- CDNA5 async global->LDS gather availability ----
#if defined(__has_builtin)
# if __has_builtin(__builtin_amdgcn_global_load_async_to_lds_b32)
#  define HAVE_ASYNC_BUILTIN 1
# endif
#endif

#if defined(HAVE_ASYNC_BUILTIN)
typedef __attribute__((address_space(1))) int gas_int;   // global
typedef __attribute__((address_space(3))) int las_int;   // LDS
#endif

__device__ __forceinline__ void wait_async49() {
  asm volatile("s_wait_asynccnt 49" ::: "memory");   // 7*(DEPTH-1)
}
__device__ __forceinline__ void wait_async0() {
  asm volatile("s_wait_asynccnt 0" ::: "memory");
}
__device__ __forceinline__ void wait_ds0() {
  asm volatile("s_wait_dscnt 0" ::: "memory");
}

// logaddexp with the same association/numerics as jnp.logaddexp
__device__ __forceinline__ float lae(float x, float y) {
  float m = fmaxf(x, y);
  float d = fminf(x, y) - m;          // <= 0, finite (NEGV is finite)
  return m + log1pf(__expf(d));
}

__global__ void ctc_zero(float* o) {
  if (threadIdx.x == 0 && blockIdx.x == 0) o[0] = 0.0f;
}

__global__ __launch_bounds__(32)
void ctc_scan(const float* __restrict__ lp,      // (T,B,V) log-probs
              const int*   __restrict__ in_len,  // (B,)
              const int*   __restrict__ tgt_all, // (B,S)
              const int*   __restrict__ tgt_len, // (B,)
              float* __restrict__ out)           // scalar
{
  __shared__ float ebuf[DEPTH][ROWF];  // async-gathered emissions, 8 timesteps deep
  __shared__ float aout[ROWF];         // final alphas for the tail reduction

  const int lane = threadIdx.x;        // wave32: one wave per batch row
  const int b    = blockIdx.x;
  const int tl   = tgt_len[b];
  const int Tin  = in_len[b];          // alpha frozen for t >= Tin -> stop there
  const int* tgt = tgt_all + b * SS;

  // ---- per-lane static state: labels, skip/valid masks ----
  int   lab[NPOS];
  bool  skipb[NPOS], validb[NPOS];
  float a[NPOS];
  #pragma unroll
  for (int j = 0; j < NPOS; ++j) {
    const int s  = lane * NPOS + j;
    const bool inr = (s < SE);
    int l = BLANKTOK;
    if (inr && (s & 1)) l = tgt[(s - 1) >> 1];
    lab[j]    = inr ? l : 0;
    skipb[j]  = inr && (s & 1) && (s >= 3) && (tgt[(s - 1) >> 1] != tgt[(s - 3) >> 1]);
    validb[j] = inr && (s < 2 * tl + 1);
  }

  const char* gbase = (const char*)lp;
#if !defined(HAVE_ASYNC_BUILTIN)
  const uint64_t gbase64 = (uint64_t)(uintptr_t)lp;
  const uint32_t ebase   = (uint32_t)(uintptr_t)&ebuf[0][0];  // LDS byte offset
#endif

  // issue 7 async b32 gathers (one emission per owned position) for timestep t
  auto issue_ts = [&](int t) {
    const int slot = t & (DEPTH - 1);
    const uint32_t row = ((uint32_t)t * BB + (uint32_t)b) * (VV * 4u);
    #pragma unroll
    for (int j = 0; j < NPOS; ++j) {
      const uint32_t goff = row + ((uint32_t)lab[j] << 2);
#if defined(HAVE_ASYNC_BUILTIN)
      __builtin_amdgcn_global_load_async_to_lds_b32(
          (gas_int*)(gbase + goff),
          (las_int*)&ebuf[slot][j * 32 + lane],
          0, 0);
#else
      const uint32_t laddr = ebase + (uint32_t)(slot * ROWF + j * 32 + lane) * 4u;
      asm volatile("global_load_async_to_lds_b32 %0, %1, %2"
                   :: "v"(laddr), "v"(goff), "s"(gbase64) : "memory");
#endif
    }
  };

  auto consume = [&](int t, float* e) {
    const int slot = t & (DEPTH - 1);
    #pragma unroll
    for (int j = 0; j < NPOS; ++j) e[j] = ebuf[slot][j * 32 + lane];
  };

  // ---- prologue: fill the pipeline (Tin >= DEPTH always here) ----
  #pragma unroll
  for (int p = 0; p < DEPTH; ++p) issue_ts(p);

  // ---- t = 0: init alpha ----
  {
    wait_async49();
    float e[NPOS];
    consume(0, e);
    wait_ds0();
    if (DEPTH < Tin) issue_ts(DEPTH);  // refill slot 0 with t=8
    #pragma unroll
    for (int j = 0; j < NPOS; ++j) {
      const int s = lane * NPOS + j;
      float v = NEGV;
      if (s == 0) v = e[j];
      else if (s == 1) v = (tl > 0) ? e[j] : NEGV;
      a[j] = validb[j] ? v : NEGV;
    }
  }

  // ---- main scan: t = 1 .. Tin-1 (alpha frozen beyond) ----
  for (int t = 1; t < Tin; ++t) {
    if (t + DEPTH <= Tin) wait_async49(); else wait_async0();

    float e[NPOS];
    consume(t, e);
    wait_ds0();                         // LDS reads landed before slot reuse
    if (t + DEPTH < Tin) issue_ts(t + DEPTH);
    if (t + 2 * DEPTH < Tin) {          // pre-warm L2: 32 lanes x 64B covers the 2KB row
      __builtin_prefetch(gbase + ((size_t)(t + 2 * DEPTH) * BB + b) * (VV * 4) + lane * 64, 0, 3);
    }

    // neighbor alphas from the previous lane (positions s-1, s-2 of 7L, 7L+1)
    float up1 = __shfl_up(a[NPOS - 1], 1);
    float up2 = __shfl_up(a[NPOS - 2], 1);
    if (lane == 0) { up1 = NEGV; up2 = NEGV; }

    float n[NPOS];
    #pragma unroll
    for (int j = 0; j < NPOS; ++j) {
      const float a1 = a[j];
      const float a2 = (j == 0) ? up1 : a[j - 1];
      float a3 = (j == 0) ? up2 : ((j == 1) ? up1 : a[j - 2]);
      a3 = skipb[j] ? a3 : NEGV;
      const float r = lae(lae(a1, a2), a3) + e[j];
      n[j] = validb[j] ? r : NEGV;
    }
    #pragma unroll
    for (int j = 0; j < NPOS; ++j) a[j] = n[j];
  }

  // ---- tail: loglik = logaddexp(alpha[2L], alpha[2L-1]); zero_infinity; sum ----
  #pragma unroll
  for (int j = 0; j < NPOS; ++j) aout[lane * NPOS + j] = a[j];
  __syncthreads();
  if (lane == 0) {
    const int last = 2 * tl;
    const float al = aout[last];
    int pidx = last - 1; if (pidx < 0) pidx = 0;
    const float ap = aout[pidx];
    const float ll = lae(al, ap);
    const float loss = (ll < 0.5f * NEGV) ? 0.0f : -ll;
    atomicAdd(out, loss);
  }
}

extern "C" void kernel_launch(void* const* d_in, const int* in_sizes, int n_in,
                              void* d_out, int out_size, void* d_ws, size_t ws_size,
                              hipStream_t stream) {
  (void)in_sizes; (void)n_in; (void)out_size; (void)d_ws; (void)ws_size;
  const float* lp  = (const float*)d_in[0];   // log_prob (T,B,V) f32
  const int*   il  = (const int*)d_in[1];     // input_lengths (B,)
  const int*   tg  = (const int*)d_in[2];     // targets (B,S)
  const int*   tll = (const int*)d_in[3];     // target_lengths (B,)
  float* out = (float*)d_out;

  ctc_zero<<<1, 32, 0, stream>>>(out);
  ctc_scan<<<BB, 32, 0, stream>>>(lp, il, tg, tll, out);
}